// PDSAC_71098888618769
// MI455X (gfx1250) — compile-verified
//
#include <hip/hip_runtime.h>
#include <math.h>

typedef __attribute__((ext_vector_type(2))) float v2f;
typedef __attribute__((ext_vector_type(8))) float v8f;
typedef __attribute__((ext_vector_type(4))) int   v4i;

#define NPTS     100000
#define NBATCH   32
#define MHYP     20
#define TILE     128
#define LDSTR    132                      // word stride: conflict-free & 16B aligned
#define NBLK     8                        // partial-gram blocks per (batch,cloud)
#define NTILES   ((NPTS + TILE - 1) / TILE)   // 782

#if __has_builtin(__builtin_amdgcn_global_load_async_to_lds_b128)
#define HAVE_ASYNC_LDS 1
#else
#define HAVE_ASYNC_LDS 0
#endif

static __device__ __forceinline__ void wait_async0()
{
#if __has_builtin(__builtin_amdgcn_s_wait_asynccnt)
    __builtin_amdgcn_s_wait_asynccnt(0);
#else
    asm volatile("s_wait_asynccnt 0x0" ::: "memory");
#endif
}

// ---------------------------------------------------------------------------
// Phase 1: partial Gram S = X X^T per (batch, cloud) via V_WMMA_F32_16X16X4_F32
// X rows: 0..2 = pc xyz, 3 = ones (zero for padding), 4..6 = apc xyz, 7..15 = 0
// Staging uses GLOBAL_LOAD_ASYNC_TO_LDS_B128 (ASYNCcnt) when available.
// ---------------------------------------------------------------------------
__global__ __launch_bounds__(32)
void gram_kernel(const float* __restrict__ A,  const float* __restrict__ Ag,
                 const float* __restrict__ B,  const float* __restrict__ Bg,
                 float* __restrict__ partial)
{
    __shared__ float X[16 * LDSTR];
    const int lane = threadIdx.x;
    const int nb   = blockIdx.x;          // 0..NBLK-1
    const int bp   = blockIdx.y;          // 0..63 : b + 32*q
    const int b    = bp & 31;
    const int q    = bp >> 5;

    const float* pc  = (q ? B  : A ) + (size_t)b * 3 * NPTS;
    const float* apc = (q ? Bg : Ag) + (size_t)b * 3 * NPTS;

    // zero rows 7..15 once
    for (int i = lane; i < 9 * LDSTR; i += 32) X[7 * LDSTR + i] = 0.0f;

    v8f acc = {0.f, 0.f, 0.f, 0.f, 0.f, 0.f, 0.f, 0.f};
    const int m    = lane & 15;           // A/B fragment row
    const int koff = (lane >> 4) << 1;    // K pair: lanes 0-15 -> K0/K1, 16-31 -> K2/K3
    const int col  = lane * 4;

    for (int t = nb; t < NTILES; t += NBLK) {
        const int t0 = t * TILE;
        __syncthreads();                  // protect previous tile's LDS reads

        const bool full = (t0 + TILE <= NPTS);

        // ---- stage 6 data rows ----
        #pragma unroll
        for (int r = 0; r < 6; ++r) {
            const float* src  = (r < 3) ? (pc + r * NPTS) : (apc + (r - 3) * NPTS);
            const int ldsrow  = (r < 3) ? r : r + 1;
            const int gi      = t0 + col;
            if (full) {
#if HAVE_ASYNC_LDS
                __builtin_amdgcn_global_load_async_to_lds_b128(
                    (v4i*)(src + gi),
                    (v4i*)&X[ldsrow * LDSTR + col], 0, 0);
#else
                const float4 v = *(const float4*)(src + gi);
                *(float4*)(&X[ldsrow * LDSTR + col]) = v;
#endif
                const int nt0 = t0 + TILE * NBLK;     // prefetch our next tile
                if (nt0 + TILE <= NPTS) __builtin_prefetch(src + nt0 + col, 0, 0);
            } else {
                float v0 = (gi + 0 < NPTS) ? src[gi + 0] : 0.f;
                float v1 = (gi + 1 < NPTS) ? src[gi + 1] : 0.f;
                float v2 = (gi + 2 < NPTS) ? src[gi + 2] : 0.f;
                float v3 = (gi + 3 < NPTS) ? src[gi + 3] : 0.f;
                X[ldsrow * LDSTR + col + 0] = v0;
                X[ldsrow * LDSTR + col + 1] = v1;
                X[ldsrow * LDSTR + col + 2] = v2;
                X[ldsrow * LDSTR + col + 3] = v3;
            }
        }
        // ---- homogeneous "ones" row (zero on padded points) ----
        {
            const int gi = t0 + col;
            float4 o;
            o.x = (gi + 0 < NPTS) ? 1.f : 0.f;
            o.y = (gi + 1 < NPTS) ? 1.f : 0.f;
            o.z = (gi + 2 < NPTS) ? 1.f : 0.f;
            o.w = (gi + 3 < NPTS) ? 1.f : 0.f;
            *(float4*)(&X[3 * LDSTR + col]) = o;
        }
#if HAVE_ASYNC_LDS
        if (full) wait_async0();
#endif
        __syncthreads();

        // ---- 32 WMMAs: acc[r][c] += sum_k X[r,k]*X[c,k] (A frag == B frag) ----
        #pragma unroll 4
        for (int kk = 0; kk < TILE; kk += 4) {
            const float2 a = *(const float2*)(&X[m * LDSTR + kk + koff]);
            v2f av; av.x = a.x; av.y = a.y;
            acc = __builtin_amdgcn_wmma_f32_16x16x4_f32(
                      false, av, false, av, (short)0, acc, false, false);
        }
    }

    float* dst = partial + ((size_t)(bp * NBLK + nb)) * 256 + lane * 8;
    #pragma unroll
    for (int v = 0; v < 8; ++v) dst[v] = acc[v];
}

// ---------------------------------------------------------------------------
// Phase 2: combine partials, per-hypothesis Kabsch (SVD via Jacobi of H^T H),
// moment-form errors, wave argmin, write best 4x4 transform per (batch,cloud)
// ---------------------------------------------------------------------------
__global__ __launch_bounds__(32)
void solve_kernel(const float* __restrict__ A,  const float* __restrict__ Ag,
                  const float* __restrict__ B,  const float* __restrict__ Bg,
                  const int* __restrict__ idxA, const int* __restrict__ idxB,
                  const float* __restrict__ partial, float* __restrict__ bestT)
{
    __shared__ float Sh[16][8];           // Sh[c][r] = D[r][c] = S(r,c)
    const int lane = threadIdx.x;
    const int bp   = blockIdx.x;
    const int b    = bp & 31;
    const int q    = bp >> 5;
    const float* pc  = (q ? B  : A ) + (size_t)b * 3 * NPTS;
    const float* apc = (q ? Bg : Ag) + (size_t)b * 3 * NPTS;
    const int*   idx = q ? idxB : idxA;

    if (lane < 16) {                      // entries with r<8 live in lanes 0..15, vgprs 0..7
        float s[8] = {0.f,0.f,0.f,0.f,0.f,0.f,0.f,0.f};
        for (int nb = 0; nb < NBLK; ++nb) {
            const float* src = partial + ((size_t)(bp * NBLK + nb)) * 256 + lane * 8;
            #pragma unroll
            for (int v = 0; v < 8; ++v) s[v] += src[v];
        }
        #pragma unroll
        for (int v = 0; v < 8; ++v) Sh[lane][v] = s[v];
    }
    __syncthreads();

    const float cogx = Sh[3][0], cogy = Sh[3][1], cogz = Sh[3][2];
    const float acx  = Sh[3][4], acy  = Sh[3][5], acz  = Sh[3][6];
    const float cog[3]  = {cogx, cogy, cogz};
    const float acog[3] = {acx,  acy,  acz };

    float T[4][4];
    float err = INFINITY;
    const int m = lane;

    if (m < MHYP) {
        // gather (reproduces torch/jax row-major .view channel mixing)
        float mp[3][4], ma[3][4];
        #pragma unroll
        for (int c = 0; c < 3; ++c)
            #pragma unroll
            for (int k = 0; k < 4; ++k) {
                const int f  = 12 * m + 4 * c + k;
                const int ch = f / 80;
                const int j  = f - 80 * ch;
                const int i  = idx[j];
                mp[c][k] = pc [ch * NPTS + i] - cog[ch];
                ma[c][k] = apc[ch * NPTS + i] - acog[ch];
            }
        // covariance H[c][d] = sum_k mp[c][k]*ma[d][k]
        float H[3][3];
        #pragma unroll
        for (int c = 0; c < 3; ++c)
            #pragma unroll
            for (int d = 0; d < 3; ++d)
                H[c][d] = mp[c][0]*ma[d][0] + mp[c][1]*ma[d][1]
                        + mp[c][2]*ma[d][2] + mp[c][3]*ma[d][3];

        // K = H^T H, Jacobi eigendecomposition -> V, eigvals
        float K[3][3], V[3][3];
        #pragma unroll
        for (int i = 0; i < 3; ++i)
            #pragma unroll
            for (int j = 0; j < 3; ++j) {
                K[i][j] = H[0][i]*H[0][j] + H[1][i]*H[1][j] + H[2][i]*H[2][j];
                V[i][j] = (i == j) ? 1.f : 0.f;
            }
        const int PQ[3][2] = {{0,1},{0,2},{1,2}};
        for (int sweep = 0; sweep < 12; ++sweep)
            #pragma unroll
            for (int r = 0; r < 3; ++r) {
                const int p = PQ[r][0], qq = PQ[r][1];
                const float apq = K[p][qq];
                if (fabsf(apq) > 1e-20f) {
                    const float tau = (K[qq][qq] - K[p][p]) / (2.f * apq);
                    const float tt  = copysignf(1.f, tau) / (fabsf(tau) + sqrtf(1.f + tau*tau));
                    const float cc  = 1.f / sqrtf(1.f + tt*tt);
                    const float ss  = tt * cc;
                    #pragma unroll
                    for (int k = 0; k < 3; ++k) {        // K <- K*G
                        const float kp = K[k][p], kq = K[k][qq];
                        K[k][p] = cc*kp - ss*kq;  K[k][qq] = ss*kp + cc*kq;
                    }
                    #pragma unroll
                    for (int k = 0; k < 3; ++k) {        // K <- G^T*K
                        const float kp = K[p][k], kq = K[qq][k];
                        K[p][k] = cc*kp - ss*kq;  K[qq][k] = ss*kp + cc*kq;
                    }
                    #pragma unroll
                    for (int k = 0; k < 3; ++k) {        // V <- V*G
                        const float vp = V[k][p], vq = V[k][qq];
                        V[k][p] = cc*vp - ss*vq;  V[k][qq] = ss*vp + cc*vq;
                    }
                }
            }
        // U[:,i] = H V[:,i] / s_i ;  rot = V U^T  (perm/sign invariant)
        float U[3][3];
        #pragma unroll
        for (int i = 0; i < 3; ++i) {
            const float si = sqrtf(fmaxf(K[i][i], 1e-30f));
            #pragma unroll
            for (int r2 = 0; r2 < 3; ++r2)
                U[r2][i] = (H[r2][0]*V[0][i] + H[r2][1]*V[1][i] + H[r2][2]*V[2][i]) / si;
        }
        #pragma unroll
        for (int a = 0; a < 3; ++a)
            #pragma unroll
            for (int d = 0; d < 3; ++d)
                T[a][d] = V[a][0]*U[d][0] + V[a][1]*U[d][1] + V[a][2]*U[d][2];
        T[0][3] = acx - cogx;  T[1][3] = acy - cogy;  T[2][3] = acz - cogz;
        T[3][0] = 0.f; T[3][1] = 0.f; T[3][2] = 0.f; T[3][3] = 1.f;

        // err = <T^T T, S_pp> - 2<T, S_ap>   (+ const, irrelevant for argmin)
        float e = 0.f;
        #pragma unroll
        for (int c = 0; c < 4; ++c)
            #pragma unroll
            for (int d = 0; d < 4; ++d) {
                const float Gcd = T[0][c]*T[0][d] + T[1][c]*T[1][d]
                                + T[2][c]*T[2][d] + T[3][c]*T[3][d];
                const float Spp = Sh[d][c];
                const float Sap = (c < 3) ? Sh[d][4 + c] : Sh[d][3];
                e += Gcd * Spp - 2.f * T[c][d] * Sap;
            }
        err = e;
    }

    // wave-wide argmin, first-minimum tie rule
    float be = err; int bi = m;
    #pragma unroll
    for (int off = 16; off > 0; off >>= 1) {
        const float oe = __shfl_xor(be, off, 32);
        const int   oi = __shfl_xor(bi, off, 32);
        if (oe < be || (oe == be && oi < bi)) { be = oe; bi = oi; }
    }
    if (m == bi && m < MHYP) {
        float* dst = bestT + (size_t)bp * 16;
        #pragma unroll
        for (int r = 0; r < 4; ++r)
            #pragma unroll
            for (int c = 0; c < 4; ++c) dst[r * 4 + c] = T[r][c];
    }
}

// ---------------------------------------------------------------------------
// Phase 3: out[b] = inv(T_A[b]) @ T_B[b]
// ---------------------------------------------------------------------------
__device__ void inv4x4(const float m[16], float o[16])
{
    float inv[16];
    inv[0]  =  m[5]*m[10]*m[15] - m[5]*m[11]*m[14] - m[9]*m[6]*m[15] + m[9]*m[7]*m[14] + m[13]*m[6]*m[11] - m[13]*m[7]*m[10];
    inv[4]  = -m[4]*m[10]*m[15] + m[4]*m[11]*m[14] + m[8]*m[6]*m[15] - m[8]*m[7]*m[14] - m[12]*m[6]*m[11] + m[12]*m[7]*m[10];
    inv[8]  =  m[4]*m[9]*m[15]  - m[4]*m[11]*m[13] - m[8]*m[5]*m[15] + m[8]*m[7]*m[13] + m[12]*m[5]*m[11] - m[12]*m[7]*m[9];
    inv[12] = -m[4]*m[9]*m[14]  + m[4]*m[10]*m[13] + m[8]*m[5]*m[14] - m[8]*m[6]*m[13] - m[12]*m[5]*m[10] + m[12]*m[6]*m[9];
    inv[1]  = -m[1]*m[10]*m[15] + m[1]*m[11]*m[14] + m[9]*m[2]*m[15] - m[9]*m[3]*m[14] - m[13]*m[2]*m[11] + m[13]*m[3]*m[10];
    inv[5]  =  m[0]*m[10]*m[15] - m[0]*m[11]*m[14] - m[8]*m[2]*m[15] + m[8]*m[3]*m[14] + m[12]*m[2]*m[11] - m[12]*m[3]*m[10];
    inv[9]  = -m[0]*m[9]*m[15]  + m[0]*m[11]*m[13] + m[8]*m[1]*m[15] - m[8]*m[3]*m[13] - m[12]*m[1]*m[11] + m[12]*m[3]*m[9];
    inv[13] =  m[0]*m[9]*m[14]  - m[0]*m[10]*m[13] - m[8]*m[1]*m[14] + m[8]*m[2]*m[13] + m[12]*m[1]*m[10] - m[12]*m[2]*m[9];
    inv[2]  =  m[1]*m[6]*m[15]  - m[1]*m[7]*m[14]  - m[5]*m[2]*m[15] + m[5]*m[3]*m[14] + m[13]*m[2]*m[7]  - m[13]*m[3]*m[6];
    inv[6]  = -m[0]*m[6]*m[15]  + m[0]*m[7]*m[14]  + m[4]*m[2]*m[15] - m[4]*m[3]*m[14] - m[12]*m[2]*m[7]  + m[12]*m[3]*m[6];
    inv[10] =  m[0]*m[5]*m[15]  - m[0]*m[7]*m[13]  - m[4]*m[1]*m[15] + m[4]*m[3]*m[13] + m[12]*m[1]*m[7]  - m[12]*m[3]*m[5];
    inv[14] = -m[0]*m[5]*m[14]  + m[0]*m[6]*m[13]  + m[4]*m[1]*m[14] - m[4]*m[2]*m[13] - m[12]*m[1]*m[6]  + m[12]*m[2]*m[5];
    inv[3]  = -m[1]*m[6]*m[11]  + m[1]*m[7]*m[10]  + m[5]*m[2]*m[11] - m[5]*m[3]*m[10] - m[9]*m[2]*m[7]   + m[9]*m[3]*m[6];
    inv[7]  =  m[0]*m[6]*m[11]  - m[0]*m[7]*m[10]  - m[4]*m[2]*m[11] + m[4]*m[3]*m[10] + m[8]*m[2]*m[7]   - m[8]*m[3]*m[6];
    inv[11] = -m[0]*m[5]*m[11]  + m[0]*m[7]*m[9]   + m[4]*m[1]*m[11] - m[4]*m[3]*m[9]  - m[8]*m[1]*m[7]   + m[8]*m[3]*m[5];
    inv[15] =  m[0]*m[5]*m[10]  - m[0]*m[6]*m[9]   - m[4]*m[1]*m[10] + m[4]*m[2]*m[9]  + m[8]*m[1]*m[6]   - m[8]*m[2]*m[5];
    const float det = m[0]*inv[0] + m[1]*inv[4] + m[2]*inv[8] + m[3]*inv[12];
    const float id  = 1.0f / det;
    #pragma unroll
    for (int i = 0; i < 16; ++i) o[i] = inv[i] * id;
}

__global__ __launch_bounds__(32)
void finish_kernel(const float* __restrict__ bestT, float* __restrict__ out)
{
    const int b = threadIdx.x;
    if (b >= NBATCH) return;
    float TA[16], TB[16], IA[16];
    #pragma unroll
    for (int i = 0; i < 16; ++i) { TA[i] = bestT[b * 16 + i]; TB[i] = bestT[(32 + b) * 16 + i]; }
    inv4x4(TA, IA);
    #pragma unroll
    for (int i = 0; i < 4; ++i)
        #pragma unroll
        for (int j = 0; j < 4; ++j)
            out[b * 16 + i * 4 + j] = IA[i*4+0]*TB[0*4+j] + IA[i*4+1]*TB[1*4+j]
                                    + IA[i*4+2]*TB[2*4+j] + IA[i*4+3]*TB[3*4+j];
}

// ---------------------------------------------------------------------------
extern "C" void kernel_launch(void* const* d_in, const int* in_sizes, int n_in,
                              void* d_out, int out_size, void* d_ws, size_t ws_size,
                              hipStream_t stream)
{
    const float* A    = (const float*)d_in[0];
    const float* Ag   = (const float*)d_in[1];
    const float* B    = (const float*)d_in[2];
    const float* Bg   = (const float*)d_in[3];
    const int*   idxA = (const int*)  d_in[4];
    const int*   idxB = (const int*)  d_in[5];
    float*       out  = (float*)d_out;

    float* partial = (float*)d_ws;                         // 64*NBLK*256 floats
    float* bestT   = partial + (size_t)64 * NBLK * 256;    // 64*16 floats

    dim3 g1(NBLK, 64, 1);
    gram_kernel<<<g1, 32, 0, stream>>>(A, Ag, B, Bg, partial);
    solve_kernel<<<64, 32, 0, stream>>>(A, Ag, B, Bg, idxA, idxB, partial, bestT);
    finish_kernel<<<1, 32, 0, stream>>>(bestT, out);
}